// TokenChoiceRouter_34462817583109
// MI455X (gfx1250) — compile-verified
//
#include <hip/hip_runtime.h>
#include <hip/hip_bf16.h>

// TokenChoiceRouter for MI455X (gfx1250).
// logits via V_WMMA_F32_16X16X4_F32, bandwidth-bound streaming of x.

typedef float v2f __attribute__((ext_vector_type(2)));
typedef float v8f __attribute__((ext_vector_type(8)));

#define RT_B   4
#define RT_T   4096
#define RT_D   2048
#define RT_TOT (RT_B * RT_T)   // 16384 tokens
#define WSTRIDE 2052           // padded W row stride in LDS (breaks bank conflicts)

// ---------------------------------------------------------------------------
// Kernel 0: zero the reduction counters in workspace (deterministic each call)
// ---------------------------------------------------------------------------
__global__ void rt_init_kernel(int* __restrict__ cnt, float* __restrict__ zsum) {
    if (threadIdx.x == 0 && blockIdx.x == 0) {
        cnt[0] = 0; cnt[1] = 0; cnt[2] = 0;
        *zsum = 0.0f;
    }
}

// ---------------------------------------------------------------------------
// Kernel 1: WMMA router.
// 256 threads = 8 waves; each wave owns a 16-token tile and runs the K=2048
// reduction as 512 V_WMMA_F32_16X16X4_F32 ops (4 accumulators, unrolled x4).
// A-matrix (16x4 f32, ISA 7.12.2): lanes 0-15 -> rows 0..15, K={kc,kc+1};
// lanes 16-31 -> rows 0..15, K={kc+2,kc+3}  => one b64 load / lane / step.
// B-matrix (4x16): column n = lane&15 from W row n (rows >=3 read a zero row).
// ---------------------------------------------------------------------------
__global__ __launch_bounds__(256) void rt_router_kernel(
    const float* __restrict__ x, const float* __restrict__ W,
    const int* __restrict__ rid_p, float* __restrict__ out,
    int* __restrict__ cnt, float* __restrict__ zsum)
{
    __shared__ float lds_w[4 * WSTRIDE];   // rows 0..2 = W, row 3 = zeros
    __shared__ float post[8][16 * 4];      // per-wave [token][logit 0..2 + pad]

    const int tid = threadIdx.x;

    // Stage W into LDS (3 real rows + zero pad row), fully coalesced.
    for (int i = tid; i < 4 * RT_D; i += 256) {
        const int row = i >> 11;          // /2048
        const int col = i & (RT_D - 1);
        lds_w[row * WSTRIDE + col] = (row < 3) ? W[row * RT_D + col] : 0.0f;
    }
    __syncthreads();

    const int lane = tid & 31;
    const int wv   = tid >> 5;
    const int half = lane >> 4;           // 0: K low pair, 1: K high pair
    const int n    = lane & 15;           // A row within tile / B column
    const int token_base = (blockIdx.x * 8 + wv) * 16;

    const float* aptr = x + (size_t)(token_base + n) * RT_D + 2 * half;
    const float* bptr = &lds_w[(n < 3 ? n : 3) * WSTRIDE + 2 * half];

    v8f c0 = {}, c1 = {}, c2 = {}, c3 = {};
    for (int kc = 0; kc < RT_D; kc += 16) {
        __builtin_prefetch(aptr + kc + 1024, 0, 0);   // global_prefetch_b8
        v2f a0 = *(const v2f*)(aptr + kc + 0);
        v2f b0 = *(const v2f*)(bptr + kc + 0);
        c0 = __builtin_amdgcn_wmma_f32_16x16x4_f32(false, a0, false, b0,
                                                   (short)0, c0, false, false);
        v2f a1 = *(const v2f*)(aptr + kc + 4);
        v2f b1 = *(const v2f*)(bptr + kc + 4);
        c1 = __builtin_amdgcn_wmma_f32_16x16x4_f32(false, a1, false, b1,
                                                   (short)0, c1, false, false);
        v2f a2 = *(const v2f*)(aptr + kc + 8);
        v2f b2 = *(const v2f*)(bptr + kc + 8);
        c2 = __builtin_amdgcn_wmma_f32_16x16x4_f32(false, a2, false, b2,
                                                   (short)0, c2, false, false);
        v2f a3 = *(const v2f*)(aptr + kc + 12);
        v2f b3 = *(const v2f*)(bptr + kc + 12);
        c3 = __builtin_amdgcn_wmma_f32_16x16x4_f32(false, a3, false, b3,
                                                   (short)0, c3, false, false);
    }
    c0 += c1; c2 += c3; c0 += c2;

    // C/D layout: VGPR r, lane L -> M = r + 8*(L>>4), N = L&15.
    // Park the 3 valid logit columns in LDS so lane m can gather token m's row.
    if (n < 3) {
#pragma unroll
        for (int r = 0; r < 8; ++r)
            post[wv][(r + 8 * half) * 4 + n] = c0[r];
    }
    asm volatile("s_wait_dscnt 0" ::: "memory");
    __syncthreads();

    if (lane < 16) {
        const int m   = lane;
        const int gid = token_base + m;
        const float l0 = post[wv][m * 4 + 0];
        const float l1 = post[wv][m * 4 + 1];
        const float l2 = post[wv][m * 4 + 2];

        // argmax (first-max tie-break, matches jnp.argmax)
        int a = 0; float bv = l0;
        if (l1 > bv) { bv = l1; a = 1; }
        if (l2 > bv) { bv = l2; a = 2; }

        // stable softmax over 3 + logsumexp
        const float mx = fmaxf(l0, fmaxf(l1, l2));
        const float e0 = __expf(l0 - mx);
        const float e1 = __expf(l1 - mx);
        const float e2 = __expf(l2 - mx);
        const float s  = e0 + e1 + e2;
        const float z  = mx + __logf(s);

        int rid = *rid_p;
        int gi  = rid < 0 ? 0 : (rid > 2 ? 2 : rid);   // min(rid, N_REC-1)
        const float cg = (gi == 0 ? e0 : (gi == 1 ? e1 : e2)) / s;

        out[2 * RT_TOT + gid] = l0;        // logits[:, :, :1]
        out[gid]              = (float)a;  // stash assign in "selected" region
        out[RT_TOT + gid]     = cg;        // stash cur_gate in "gate_weights" region

        atomicAdd(&cnt[a], 1);
        atomicAdd(zsum, z * z);
    }
}

// ---------------------------------------------------------------------------
// Kernel 2: stable compaction, one block per batch row (exact _compact).
// Reads the stashed assign (region 0) / cur_gate (region 1) and overwrites
// them in place with sel / gv.  Scatter positions in chunk k are always
// < k-chunk end <= first index read by chunk k+1, so no RAW hazard.
// ---------------------------------------------------------------------------
__global__ __launch_bounds__(1024) void rt_compact_kernel(
    const int* __restrict__ rid_p, float* __restrict__ out)
{
    const int b    = blockIdx.x;
    const int tid  = threadIdx.x;
    const int lane = tid & 31;
    const int wid  = tid >> 5;

    float* selbuf = out + (size_t)b * RT_T;            // region 0 slice
    float* gvbuf  = out + RT_TOT + (size_t)b * RT_T;   // region 1 slice

    __shared__ int   wsum[32];
    __shared__ int   wpre[32];
    __shared__ int   sbase;
    __shared__ int   slast_sel;
    __shared__ float slast_gv;
    if (tid == 0) { sbase = 0; slast_sel = 0; slast_gv = 0.0f; }

    const float ridf = (float)(*rid_p);

    for (int t0 = 0; t0 < RT_T; t0 += 1024) {
        const int   t    = t0 + tid;
        const float aval = selbuf[t];   // stashed assign (0/1/2 as float)
        const float cg   = gvbuf[t];    // stashed cur_gate
        const int active = (aval >= ridf) ? 1 : 0;
        __syncthreads();                // all chunk reads done before writes

        const unsigned mask   = (unsigned)__ballot(active);
        const int      prefix = __popc(mask & ((1u << lane) - 1u));
        if (lane == 0) wsum[wid] = __popc(mask);
        __syncthreads();
        if (tid == 0) {
            int run = sbase;
            for (int i = 0; i < 32; ++i) { wpre[i] = run; run += wsum[i]; }
            sbase = run;
        }
        __syncthreads();
        const int chunk_end = sbase;    // total count through this chunk
        if (active) {
            const int pos = wpre[wid] + prefix;
            selbuf[pos] = (float)t;
            gvbuf[pos]  = cg;
            if (pos == chunk_end - 1) { slast_sel = t; slast_gv = cg; }
        }
        __syncthreads();                // protect wsum/wpre/slast reuse
    }

    const int   count = sbase;
    const int   ls    = slast_sel;
    const float lg    = slast_gv;
    for (int j = count + tid; j < RT_T; j += 1024) {
        selbuf[j] = (float)ls;          // pad with last active entry (or 0)
        gvbuf[j]  = lg;
    }
}

// ---------------------------------------------------------------------------
// Kernel 3: scalar losses
// ---------------------------------------------------------------------------
__global__ void rt_finalize_kernel(const int* __restrict__ cnt,
                                   const float* __restrict__ zsum,
                                   float* __restrict__ out)
{
    if (threadIdx.x == 0 && blockIdx.x == 0) {
        const float inv   = 1.0f / (float)RT_TOT;
        const float third = 1.0f / 3.0f;
        float aux = 0.0f;
        for (int i = 0; i < 3; ++i) {
            const float f = (float)cnt[i] * inv - third;
            aux += f * f;
        }
        out[3 * RT_TOT + 0] = 0.01f * aux;            // aux_loss
        out[3 * RT_TOT + 1] = 0.001f * (*zsum * inv); // z_loss
    }
}

// ---------------------------------------------------------------------------
extern "C" void kernel_launch(void* const* d_in, const int* in_sizes, int n_in,
                              void* d_out, int out_size, void* d_ws, size_t ws_size,
                              hipStream_t stream)
{
    const float* x   = (const float*)d_in[0];   // [4,4096,2048] f32
    const float* W   = (const float*)d_in[1];   // [3,2048] f32
    const int*   rid = (const int*)d_in[2];     // scalar int
    float* out = (float*)d_out;                 // 3*16384 + 2 floats

    int*   cnt  = (int*)d_ws;                   // 3 ints
    float* zsum = (float*)((char*)d_ws + 12);   // 1 float

    rt_init_kernel<<<1, 64, 0, stream>>>(cnt, zsum);
    rt_router_kernel<<<RT_TOT / (16 * 8), 256, 0, stream>>>(x, W, rid, out, cnt, zsum);
    rt_compact_kernel<<<RT_B, 1024, 0, stream>>>(rid, out);
    rt_finalize_kernel<<<1, 64, 0, stream>>>(cnt, zsum, out);
}